// TreeEncoder_30648886624756
// MI455X (gfx1250) — compile-verified
//
#include <hip/hip_runtime.h>
#include <hip/hip_bf16.h>
#include <stdint.h>

#define EMB 128
#define ROWS_PER_BLOCK 512

typedef __attribute__((ext_vector_type(2))) float v2f;
typedef __attribute__((ext_vector_type(4))) float v4f;
typedef __attribute__((ext_vector_type(8))) float v8f;

__device__ __forceinline__ unsigned int encode_f32(float f) {
  unsigned int b = __float_as_uint(f);
  return b ^ ((unsigned int)(((int)b) >> 31) | 0x80000000u);
}
__device__ __forceinline__ float decode_f32(unsigned int k) {
  unsigned int b = k ^ (((k >> 31) == 0u) ? 0xFFFFFFFFu : 0x80000000u);
  return __uint_as_float(b);
}

// ---------------- kernel 0: init encoded max accumulator ----------------
__global__ void tree_init_kernel(unsigned int* __restrict__ maxkey) {
  maxkey[threadIdx.x] = encode_f32(-INFINITY);
}

// ---------------- kernel 1: per-block embedding sums --------------------
__global__ void tree_esum_kernel(const int* __restrict__ tokens,
                                 const float* __restrict__ E,
                                 float* __restrict__ esum, int n) {
  const int b = blockIdx.x;
  const int j = threadIdx.x;
  const int r0 = b * ROWS_PER_BLOCK;
  const int r1 = min(r0 + ROWS_PER_BLOCK, n);
  float acc = 0.f;
  for (int i = r0; i < r1; ++i) {
    const int t = tokens[i];
    acc += E[(size_t)t * EMB + j];
  }
  esum[(size_t)b * EMB + j] = acc;
}

// ---------------- kernel 2: exclusive scan of block sums ----------------
__global__ void tree_escan_kernel(const float* __restrict__ esum,
                                  float* __restrict__ eprefix, int nb) {
  const int j = threadIdx.x;
  float acc = 0.f;
  for (int b = 0; b < nb; ++b) {
    eprefix[(size_t)b * EMB + j] = acc;  // exclusive
    acc += esum[(size_t)b * EMB + j];
  }
}

// ---------------- kernel 3: per-block S offset = eprefix@W^T + cnt*b ----
__global__ void tree_xoff_kernel(const float* __restrict__ eprefix,
                                 const float* __restrict__ W,
                                 const float* __restrict__ bias,
                                 float* __restrict__ xoff) {
  const int b = blockIdx.x;
  const int j = threadIdx.x;
  __shared__ float ep[EMB];
  ep[j] = eprefix[(size_t)b * EMB + j];
  __syncthreads();
  float acc = (float)((long long)b * ROWS_PER_BLOCK) * bias[j];
  for (int k = 0; k < EMB; ++k)
    acc = fmaf(ep[k], W[(size_t)j * EMB + k], acc);
  xoff[(size_t)b * EMB + j] = acc;
}

// ---------------- kernel 4: gather + fp32 WMMA GEMM + scan + store S ----
__global__ void __launch_bounds__(256, 1)
tree_gemm_scan_kernel(const int* __restrict__ tokens,
                      const float* __restrict__ E,
                      const float* __restrict__ W,
                      const float* __restrict__ bias,
                      const float* __restrict__ xoff,
                      float* __restrict__ S, int n) {
  __shared__ float Atile[16 * EMB];  // gathered embedding tile (A)
  __shared__ float Xtile[16 * EMB];  // GEMM output tile (x rows)

  const int b = blockIdx.x;
  const int tid = (int)threadIdx.x;
  const int wave = tid >> 5;   // 0..7 : output column tile (16 cols each)
  const int lane = tid & 31;
  const int r0 = b * ROWS_PER_BLOCK;
  const int rend = min(r0 + ROWS_PER_BLOCK, n);

  // ---- Preload B fragments for all 32 K-steps (held in VGPRs) ----
  // f32 WMMA 16x16x4: B(4x16).  Lane layout (transpose of A):
  //   lanes 0-15 : (K=4t+0, 4t+1), N = lane
  //   lanes 16-31: (K=4t+2, 4t+3), N = lane-16
  // B[k][n] = W^T[k][n] = W[ncol][k]
  v2f Bf[32];
  {
    const int ncol = (wave << 4) + (lane & 15);
    const int khalf = (lane >> 4) << 1;  // 0 or 2
    const float* wrow = W + (size_t)ncol * EMB;
#pragma unroll
    for (int t = 0; t < 32; ++t) {
      Bf[t].x = wrow[4 * t + khalf + 0];
      Bf[t].y = wrow[4 * t + khalf + 1];
    }
  }
  const float bj = bias[(wave << 4) + (lane & 15)];

  // Column-scan accumulator: threads 0..127 each own one column.
  float acc = 0.f;
  if (tid < EMB) acc = xoff[(size_t)b * EMB + tid];

  for (int c0 = r0; c0 < rend; c0 += 16) {
    // ---- stage gathered A tile (16 rows x 128) into LDS ----
    // Out-of-range rows are clamped (never consumed downstream), which
    // lets us issue two unconditional global_load_b128 per thread.
    {
      const int r = tid & 15;    // row within tile
      const int seg = tid >> 4;  // 0..15, 8 floats each
      const int row = min(c0 + r, n - 1);
      const int t = tokens[row];
      const v4f* src = (const v4f*)(E + (size_t)t * EMB + seg * 8);
      v4f v0 = src[0];
      v4f v1 = src[1];
      v4f* dst = (v4f*)&Atile[r * EMB + seg * 8];
      dst[0] = v0;
      dst[1] = v1;
    }
    __syncthreads();

    // ---- D(16x16) = A(16x128) x W^T(128x16) via 32 f32 WMMAs ----
    v8f C = {};
    {
      const int m = lane & 15;
      const int khalf = (lane >> 4) << 1;  // 0 or 2
      const float* arow = &Atile[m * EMB + khalf];
#pragma unroll
      for (int t = 0; t < 32; ++t) {
        v2f A;
        A.x = arow[4 * t + 0];
        A.y = arow[4 * t + 1];
        C = __builtin_amdgcn_wmma_f32_16x16x4_f32(
            /*neg_a=*/false, A, /*neg_b=*/false, Bf[t],
            /*c_mod=*/(short)0, C, /*reuse_a=*/false, /*reuse_b=*/false);
      }
    }

    // ---- spill D (+bias) to Xtile: VGPR v holds rows v (lanes<16) / v+8 ----
    {
      const int col = (wave << 4) + (lane & 15);
      const int rbase = (lane >> 4) << 3;  // 0 or 8
#pragma unroll
      for (int vv = 0; vv < 8; ++vv)
        Xtile[(vv + rbase) * EMB + col] = C[vv] + bj;
    }
    __syncthreads();

    // ---- sequential inclusive scan over 16 rows, store S ----
    if (tid < EMB) {
      float a = acc;
#pragma unroll
      for (int r = 0; r < 16; ++r) {
        const int row = c0 + r;
        if (row < rend) {
          a += Xtile[r * EMB + tid];
          S[(size_t)row * EMB + tid] = a;
        }
      }
      acc = a;
    }
    __syncthreads();
  }
}

// ---------------- kernel 5: h = S[end] - S[i-1]; column max --------------
__global__ void tree_hmax_kernel(const int* __restrict__ subtree_end,
                                 const float* __restrict__ S,
                                 unsigned int* __restrict__ maxkey, int n) {
  const int j = (int)threadIdx.x & 127;
  const int half = (int)threadIdx.x >> 7;  // 0 or 1
  float m = -INFINITY;
  for (int i = blockIdx.x * 2 + half; i < n; i += gridDim.x * 2) {
    const int e = subtree_end[i];
    float hv = S[(size_t)e * EMB + j];
    if (i > 0) hv -= S[(size_t)(i - 1) * EMB + j];
    m = fmaxf(m, hv);
  }
  __shared__ float red[2][EMB];
  red[half][j] = m;
  __syncthreads();
  if (half == 0) {
    m = fmaxf(red[0][j], red[1][j]);
    atomicMax(&maxkey[j], encode_f32(m));
  }
}

// ---------------- kernel 6: decode to output ----------------------------
__global__ void tree_finish_kernel(const unsigned int* __restrict__ maxkey,
                                   float* __restrict__ out) {
  out[threadIdx.x] = decode_f32(maxkey[threadIdx.x]);
}

extern "C" void kernel_launch(void* const* d_in, const int* in_sizes, int n_in,
                              void* d_out, int out_size, void* d_ws,
                              size_t ws_size, hipStream_t stream) {
  const int* tokens = (const int*)d_in[0];
  const int* subtree_end = (const int*)d_in[1];
  const float* E = (const float*)d_in[2];
  const float* W = (const float*)d_in[3];
  const float* bias = (const float*)d_in[4];
  float* out = (float*)d_out;

  const int n = in_sizes[0];
  const int nb = (n + ROWS_PER_BLOCK - 1) / ROWS_PER_BLOCK;

  // workspace layout
  char* ws = (char*)d_ws;
  float* S = (float*)ws;
  size_t off = (size_t)n * EMB * sizeof(float);
  float* esum = (float*)(ws + off);      off += (size_t)nb * EMB * sizeof(float);
  float* eprefix = (float*)(ws + off);   off += (size_t)nb * EMB * sizeof(float);
  float* xoff = (float*)(ws + off);      off += (size_t)nb * EMB * sizeof(float);
  unsigned int* maxkey = (unsigned int*)(ws + off);

  tree_init_kernel<<<1, EMB, 0, stream>>>(maxkey);
  tree_esum_kernel<<<nb, EMB, 0, stream>>>(tokens, E, esum, n);
  tree_escan_kernel<<<1, EMB, 0, stream>>>(esum, eprefix, nb);
  tree_xoff_kernel<<<nb, EMB, 0, stream>>>(eprefix, W, bias, xoff);
  tree_gemm_scan_kernel<<<nb, 256, 0, stream>>>(tokens, E, W, bias, xoff, S, n);
  tree_hmax_kernel<<<4096, 256, 0, stream>>>(subtree_end, S, maxkey, n);
  tree_finish_kernel<<<1, EMB, 0, stream>>>(maxkey, out);
}